// LinearAttention_30734785970263
// MI455X (gfx1250) — compile-verified
//
#include <hip/hip_runtime.h>
#include <math.h>

typedef __attribute__((ext_vector_type(2))) float v2f;
typedef __attribute__((ext_vector_type(8))) float v8f;

#define Bb 4
#define Hh 16
#define Ss 4096
#define Dd 128
#define BH (Bb * Hh)
#define CHUNKS 32
#define ROWS_PER_CHUNK (Ss / CHUNKS)   // 128
#define NW 4                           // waves per block in phase 2
#define LDS_STRIDE 132                 // 128 + 4 pad: bank-conflict-free A-fragment reads
#define EPS 1e-6f

__device__ __forceinline__ float elu1(float x) {
    // elu(x) + 1 : x>0 ? x+1 : exp(x)
    return x > 0.0f ? x + 1.0f : __expf(x);
}

// ---------------- Phase 1a: per-(bh,chunk) partial Ksum[128] and KV scalar ----------------
__global__ void __launch_bounds__(256) la_phase1a(const float* __restrict__ K,
                                                  const float* __restrict__ V,
                                                  float* __restrict__ pksum,
                                                  float* __restrict__ pkv) {
    const int chunk = blockIdx.x;
    const int bh    = blockIdx.y;
    const int t     = threadIdx.x;
    const int d     = t & (Dd - 1);
    const int half  = t >> 7;   // two rows in flight across 256 threads

    const long base = ((long)bh * Ss + (long)chunk * ROWS_PER_CHUNK) * Dd;
    float ks = 0.0f, kv = 0.0f;
#pragma unroll 4
    for (int it = 0; it < ROWS_PER_CHUNK / 2; ++it) {
        long idx = base + (long)(it * 2 + half) * Dd + d;   // coalesced 512B per half-block
        float kf = elu1(K[idx]);
        ks += kf;
        kv += kf * V[idx];
    }

    __shared__ float red[256];
    red[t] = ks;
    __syncthreads();
    if (t < Dd) pksum[((long)chunk * BH + bh) * Dd + t] = red[t] + red[t + 128];
    __syncthreads();
    red[t] = kv;
    __syncthreads();
    for (int off = 128; off > 0; off >>= 1) {               // deterministic tree
        if (t < off) red[t] += red[t + off];
        __syncthreads();
    }
    if (t == 0) pkv[chunk * BH + bh] = red[0];
}

// ---------------- Phase 1b: fold chunk partials (fixed order => deterministic) ----------------
__global__ void __launch_bounds__(128) la_phase1b(const float* __restrict__ pksum,
                                                  const float* __restrict__ pkv,
                                                  float* __restrict__ ksumF,
                                                  float* __restrict__ kvF) {
    const int bh = blockIdx.x;
    const int t  = threadIdx.x;
    float s = 0.0f;
    for (int c = 0; c < CHUNKS; ++c) s += pksum[((long)c * BH + bh) * Dd + t];
    ksumF[bh * Dd + t] = s;
    if (t == 0) {
        float kv = 0.0f;
        for (int c = 0; c < CHUNKS; ++c) kv += pkv[c * BH + bh];
        kvF[bh] = kv;
    }
}

// ---------------- Phase 2: stream Q once; WMMA f32 16x16x4 computes 16 row-dots/wave ----------------
__global__ void __launch_bounds__(128) la_phase2(const float* __restrict__ Q,
                                                 const float* __restrict__ ksumF,
                                                 const float* __restrict__ kvF,
                                                 float* __restrict__ Out) {
    __shared__ float lds_ksum[Dd];
    __shared__ float lds_qf[NW][16][LDS_STRIDE];
    __shared__ float lds_scale[NW][16];

    const int bh      = blockIdx.y;
    const int t       = threadIdx.x;
    const int wave    = t >> 5;
    const int lane    = t & 31;
    const int rowbase = blockIdx.x * (NW * 16) + wave * 16;

    if (t < Dd) lds_ksum[t] = ksumF[bh * Dd + t];

    // Load 16x128 Q tile -> LDS as elu(Q)+1 (coalesced float4, read Q exactly once)
    const float4* Q4 = (const float4*)Q;
    const long tilebase4 = ((long)bh * Ss + rowbase) * (Dd / 4);
#pragma unroll
    for (int r = 0; r < 16; ++r) {
        float4 q = Q4[tilebase4 + (long)r * (Dd / 4) + lane];
        q.x = elu1(q.x); q.y = elu1(q.y); q.z = elu1(q.z); q.w = elu1(q.w);
        *(float4*)&lds_qf[wave][r][lane * 4] = q;
    }
    __syncthreads();

    // A fragment: lane L holds row L&15, K = 4*kb + 2*(L>=16) + {0,1}
    // B fragment: Ksum broadcast across all 16 N columns -> same per-lane K slice
    const int colsel = (lane >> 4) << 1;   // 0 or 2
    const int rowA   = lane & 15;
    v8f acc = {0.f, 0.f, 0.f, 0.f, 0.f, 0.f, 0.f, 0.f};
#pragma unroll
    for (int kb = 0; kb < Dd / 4; ++kb) {
        v2f a2 = *(const v2f*)&lds_qf[wave][rowA][kb * 4 + colsel];
        v2f b2 = *(const v2f*)&lds_ksum[kb * 4 + colsel];
        acc = __builtin_amdgcn_wmma_f32_16x16x4_f32(
            /*neg_a=*/false, a2, /*neg_b=*/false, b2,
            /*c_mod=*/(short)0, acc, /*reuse_a=*/false, /*reuse_b=*/false);
    }

    // D layout: lanes 0-15 -> rows 0-7 in acc[0..7] (N=lane); lanes 16-31 -> rows 8-15
    const float kv = kvF[bh];
    if (lane == 0) {
#pragma unroll
        for (int r = 0; r < 8; ++r) lds_scale[wave][r] = kv / (acc[r] + EPS);
    } else if (lane == 16) {
#pragma unroll
        for (int r = 0; r < 8; ++r) lds_scale[wave][8 + r] = kv / (acc[r] + EPS);
    }
    __syncthreads();

    // out[row, d] = Qf[row, d] * KV / (dot + eps)  -- coalesced 512B/wave stores
    float4* Out4 = (float4*)Out;
#pragma unroll
    for (int r = 0; r < 16; ++r) {
        float sc = lds_scale[wave][r];
        float4 q = *(const float4*)&lds_qf[wave][r][lane * 4];
        float4 o;
        o.x = q.x * sc; o.y = q.y * sc; o.z = q.z * sc; o.w = q.w * sc;
        Out4[tilebase4 + (long)r * (Dd / 4) + lane] = o;
    }
}

extern "C" void kernel_launch(void* const* d_in, const int* in_sizes, int n_in,
                              void* d_out, int out_size, void* d_ws, size_t ws_size,
                              hipStream_t stream) {
    const float* Q = (const float*)d_in[0];
    const float* K = (const float*)d_in[1];
    const float* V = (const float*)d_in[2];
    float* out = (float*)d_out;

    // Workspace layout (~1.06 MB of floats), fully overwritten every call:
    float* pksum = (float*)d_ws;                        // [CHUNKS][BH][128]
    float* pkv   = pksum + (long)CHUNKS * BH * Dd;      // [CHUNKS][BH]
    float* ksumF = pkv + CHUNKS * BH;                   // [BH][128]
    float* kvF   = ksumF + BH * Dd;                     // [BH]

    la_phase1a<<<dim3(CHUNKS, BH), 256, 0, stream>>>(K, V, pksum, pkv);
    la_phase1b<<<dim3(BH), 128, 0, stream>>>(pksum, pkv, ksumF, kvF);
    la_phase2<<<dim3(Ss / (NW * 16), BH), 128, 0, stream>>>(Q, ksumF, kvF, out);
}